// SampleAndGroup_26946624815127
// MI455X (gfx1250) — compile-verified
//
#include <hip/hip_runtime.h>

// ---------------------------------------------------------------------------
// PointNet++ sample-and-group for MI455X (gfx1250, wave32).
//   out = ( neighbor[8,8,256,256] (as float), centroids[8,256,3] )
// Kernel 1: FPS (serial-in-iteration, parallel-in-points, LDS argmax).
// Kernel 2: cross ball query using V_WMMA_F32_16X16X4_F32, 2 WMMAs (32 points)
//   per iteration, global_prefetch_b8 lookahead, LDS-staged rows flushed as
//   coalesced b128 stores.
//   d2 = A x B + p2 with A-row(m) = (-2cx,-2cy,-2cz,|c|^2), B-col(n)=(x,y,z,1)
// ---------------------------------------------------------------------------

#define BB        8
#define NPTS      8192
#define NCENT     256
#define NSAMP     256
#define RAD2      0.04f      // 0.2^2
#define PSTRIDE   6          // points are [B, N, 6]; coords = first 3

typedef __attribute__((ext_vector_type(2))) float v2f;
typedef __attribute__((ext_vector_type(8))) float v8f;

// ------------------------- Kernel 1: FPS -----------------------------------
__global__ __launch_bounds__(256) void fps_kernel(const float* __restrict__ points,
                                                  float* __restrict__ out_centroids, // [B,256,3]
                                                  float* __restrict__ ws_cent)       // [B,256,4] padded
{
    const int b   = blockIdx.x;
    const int tid = threadIdx.x;

    __shared__ float s_mind2[NPTS];   // 32 KB of the 320 KB WGP LDS pool
    __shared__ float s_rv[256];
    __shared__ int   s_ri[256];
    __shared__ int   s_far;
    __shared__ float s_c[3];

    const float* pb = points + (size_t)b * NPTS * PSTRIDE;

    for (int i = tid; i < NPTS; i += 256) s_mind2[i] = 3.402823466e+38f;
    if (tid == 0) s_far = 0;
    __syncthreads();

    for (int it = 0; it < NCENT; ++it) {
        const int far = s_far;
        if (tid == 0) {
            const float cx = pb[(size_t)far * PSTRIDE + 0];
            const float cy = pb[(size_t)far * PSTRIDE + 1];
            const float cz = pb[(size_t)far * PSTRIDE + 2];
            s_c[0] = cx; s_c[1] = cy; s_c[2] = cz;
            float* oc = out_centroids + ((size_t)b * NCENT + it) * 3;
            oc[0] = cx; oc[1] = cy; oc[2] = cz;
            float* wcp = ws_cent + ((size_t)b * NCENT + it) * 4;
            wcp[0] = cx; wcp[1] = cy; wcp[2] = cz;
            wcp[3] = cx * cx + cy * cy + cz * cz;
        }
        __syncthreads();
        const float cx = s_c[0], cy = s_c[1], cz = s_c[2];

        float bv = -1.0f;
        int   bi = 0;
        for (int i = tid; i < NPTS; i += 256) {
            const float dx = pb[(size_t)i * PSTRIDE + 0] - cx;
            const float dy = pb[(size_t)i * PSTRIDE + 1] - cy;
            const float dz = pb[(size_t)i * PSTRIDE + 2] - cz;
            const float d2 = dx * dx + dy * dy + dz * dz;
            const float m  = fminf(s_mind2[i], d2);
            s_mind2[i] = m;
            if (m > bv || (m == bv && i < bi)) { bv = m; bi = i; }
        }
        s_rv[tid] = bv; s_ri[tid] = bi;
        __syncthreads();
        // first-occurrence argmax (matches jnp.argmax tie-break)
        for (int off = 128; off > 0; off >>= 1) {
            if (tid < off) {
                const float ov = s_rv[tid + off];
                const int   oi = s_ri[tid + off];
                if (ov > s_rv[tid] || (ov == s_rv[tid] && oi < s_ri[tid])) {
                    s_rv[tid] = ov; s_ri[tid] = oi;
                }
            }
            __syncthreads();
        }
        if (tid == 0) s_far = s_ri[0];
        __syncthreads();
    }
}

// ---------------- Kernel 2: WMMA ball query + ordered selection ------------
// One wave32 per 16-centroid tile; 2 WMMAs (32 points) per iteration.
// Rows staged in LDS, flushed as coalesced 128-bit stores (16 rows of a tile
// are contiguous in the global neighbor tensor).
__global__ __launch_bounds__(32) void ballq_kernel(const float* __restrict__ points,
                                                   const float* __restrict__ ws_cent,
                                                   float* __restrict__ out_nb)
{
    const int stile = blockIdx.x;        // 0..15 : tile of 16 centroids
    const int b2    = blockIdx.y;        // centroid batch
    const int b1    = blockIdx.z;        // point-cloud batch
    const int lane  = threadIdx.x;       // 0..31
    const int m     = lane & 15;         // local centroid / point column
    const bool hi   = lane >= 16;

    __shared__ float s_row[16 * NSAMP];  // 16 KB: one output row per centroid

    // A-matrix, 16x4 f32 layout: lanes 0-15 hold K=0,1; lanes 16-31 hold K=2,3.
    // Row m = (-2cx, -2cy, -2cz, |c|^2)  ->  A*B = |c|^2 - 2 c.p
    const float* wcp = ws_cent + ((size_t)b2 * NCENT + stile * 16 + m) * 4;
    const float cx = wcp[0], cy = wcp[1], cz = wcp[2], c2 = wcp[3];
    v2f a;
    a.x = hi ? (-2.0f * cz) : (-2.0f * cx);
    a.y = hi ? c2           : (-2.0f * cy);

    const float* pb = points + (size_t)b1 * NPTS * PSTRIDE;

    unsigned cnt      = 0;
    int      firstIdx = NPTS;            // pad value when no point in radius

    for (int t0 = 0; t0 < NPTS; t0 += 32) {
        // Two 16-point B tiles: col n = (x, y, z, 1).
        const float* pp0 = pb + (size_t)(t0 + m)      * PSTRIDE;
        const float* pp1 = pb + (size_t)(t0 + 16 + m) * PSTRIDE;
        const float px0 = pp0[0], py0 = pp0[1], pz0 = pp0[2];
        const float px1 = pp1[0], py1 = pp1[1], pz1 = pp1[2];

        // Prefetch ~4 iterations ahead (lowers to global_prefetch_b8).
        {
            int npf = t0 + 128 + m;
            if (npf > NPTS - 1) npf = NPTS - 1;
            __builtin_prefetch(pb + (size_t)npf * PSTRIDE, 0, 3);
        }

        const float p2a = px0 * px0 + py0 * py0 + pz0 * pz0;
        const float p2b = px1 * px1 + py1 * py1 + pz1 * pz1;
        v2f bm0, bm1;
        bm0.x = hi ? pz0  : px0;  bm0.y = hi ? 1.0f : py0;
        bm1.x = hi ? pz1  : px1;  bm1.y = hi ? 1.0f : py1;

        const v8f cz8 = {0.f, 0.f, 0.f, 0.f, 0.f, 0.f, 0.f, 0.f};
        v8f acc0 = __builtin_amdgcn_wmma_f32_16x16x4_f32(
            false, a, false, bm0, (short)0, cz8, false, false);
        v8f acc1 = __builtin_amdgcn_wmma_f32_16x16x4_f32(
            false, a, false, bm1, (short)0, cz8, false, false);

        // acc*[r]: centroid (r + (hi?8:0)). Ballot r -> bits[15:0] = centroid r,
        // bits[31:16] = centroid r+8; bit position = point column.
        unsigned m1 = 0u, m2 = 0u;
#pragma unroll
        for (int r = 0; r < 8; ++r) {
            const unsigned b1v =
                __builtin_amdgcn_ballot_w32((bool)(acc0[r] + p2a <= RAD2));
            const unsigned b2v =
                __builtin_amdgcn_ballot_w32((bool)(acc1[r] + p2b <= RAD2));
            if ((lane & 7) == r) { m1 = b1v; m2 = b2v; }
        }
        // 32-bit in-radius mask for this lane's centroid, bit k = point t0+k.
        unsigned mask32 = hi ? 0u
                             : ((lane < 8)
                                    ? ((m1 & 0xFFFFu) | (m2 << 16))
                                    : ((m1 >> 16) | (m2 & 0xFFFF0000u)));

        // append in-radius indices in ascending order (PointNet++ first-found)
        while (mask32 && cnt < NSAMP) {
            const int bit = __builtin_ctz(mask32);
            mask32 &= mask32 - 1u;
            const int idx = t0 + bit;
            if (cnt == 0) firstIdx = idx;
            s_row[m * NSAMP + cnt] = (float)idx;
            ++cnt;
        }

        const unsigned done =
            __builtin_amdgcn_ballot_w32((bool)(hi || cnt >= NSAMP));
        if (done == 0xFFFFFFFFu) break;   // uniform early exit
    }

    if (!hi) {
        const float pad = (float)firstIdx;   // first hit, or 8192 if empty
        for (unsigned k = cnt; k < NSAMP; ++k) s_row[m * NSAMP + k] = pad;
    }
    __syncthreads();

    // Coalesced flush: the 16 rows of this tile are contiguous in out_nb.
    float* dst = out_nb +
        (((size_t)(b1 * BB + b2) * NCENT) + (size_t)stile * 16) * (size_t)NSAMP;
    const float4* src4 = (const float4*)s_row;
    float4*       dst4 = (float4*)dst;
    for (int i = lane; i < (16 * NSAMP) / 4; i += 32) dst4[i] = src4[i];
}

// --------------------------------------------------------------------------
extern "C" void kernel_launch(void* const* d_in, const int* in_sizes, int n_in,
                              void* d_out, int out_size, void* d_ws, size_t ws_size,
                              hipStream_t stream)
{
    (void)in_sizes; (void)n_in; (void)out_size; (void)ws_size;
    const float* points = (const float*)d_in[0];        // [8, 8192, 6]
    float* out          = (float*)d_out;
    float* out_cent     = out + (size_t)BB * BB * NCENT * NSAMP; // centroid tail
    float* ws_cent      = (float*)d_ws;                 // [8, 256, 4] padded

    fps_kernel<<<dim3(BB), dim3(256), 0, stream>>>(points, out_cent, ws_cent);
    ballq_kernel<<<dim3(16, BB, BB), dim3(32), 0, stream>>>(points, ws_cent, out);
}